// EdgeVar_32220844654986
// MI455X (gfx1250) — compile-verified
//
#include <hip/hip_runtime.h>
#include <stdint.h>

// EdgeVar: per-edge (||p[d]-p[s]|| - 1)^2, segment-mean over 128 graphs keyed by
// batch[s], then mean over graphs -> one fp32 scalar.
//
// Memory-bound: 25.6 MB edge-index stream (HBM), random 8B gathers from an
// 800KB position table + 400KB batch table (L2-resident). CDNA5 path used:
// async global->LDS streaming of the edge-index tiles (ASYNCcnt), LDS float
// atomics for per-block 128-bin privatization, one global fp32 atomic flush
// per bin per block.
//
// NOTE: edgevar_main is intentionally the first kernel in this file so the
// compile-loop disasm snippet shows its body (async_to_lds / s_wait_asynccnt).

#define N_GRAPHS   128
#define BLOCK      256          // 8 waves of 32 on gfx1250
#define TILE_EDGES 1024         // BLOCK lanes x 4 int32 (one b128 per lane per row)

#if defined(__gfx1250__) && \
    __has_builtin(__builtin_amdgcn_global_load_async_to_lds_b128) && \
    __has_builtin(__builtin_amdgcn_s_wait_asynccnt)
#define USE_ASYNC 1
#else
#define USE_ASYNC 0
#endif

#if USE_ASYNC
typedef int v4i __attribute__((ext_vector_type(4)));
typedef __attribute__((address_space(1))) v4i glb_v4i;
typedef __attribute__((address_space(3))) v4i lds_v4i;
__device__ __forceinline__ void async_b128(const int* gptr, int* lptr) {
  // each lane copies 16B global -> LDS, tracked with ASYNCcnt
  __builtin_amdgcn_global_load_async_to_lds_b128(
      (glb_v4i*)gptr, (lds_v4i*)lptr, /*offset=*/0, /*cpol=*/0);
}
#endif

__device__ __forceinline__ void edge_accum(const float* __restrict__ pos,
                                           const int* __restrict__ batch,
                                           int s, int d,
                                           float* s_sum, float* s_cnt) {
  float2 ps = ((const float2*)pos)[s];   // global_load_b64, L2-resident gather
  float2 pd = ((const float2*)pos)[d];
  float dx = pd.x - ps.x;
  float dy = pd.y - ps.y;
  float eu = sqrtf(__fmaf_rn(dx, dx, dy * dy));
  float t  = eu - 1.0f;
  int   g  = batch[s];
  unsafeAtomicAdd(&s_sum[g], t * t);     // ds_add_f32 (LDS-privatized bins)
  unsafeAtomicAdd(&s_cnt[g], 1.0f);
}

__global__ __launch_bounds__(BLOCK) void edgevar_main(
    const float* __restrict__ pos,   // [N,2]
    const int*   __restrict__ eidx,  // [2,E] row-major: src row then dst row
    const int*   __restrict__ batch, // [N]
    float* __restrict__ ws,          // ws[0:128]=sums, ws[128:256]=counts
    int n_edges) {
  __shared__ float s_sum[N_GRAPHS];
  __shared__ float s_cnt[N_GRAPHS];
  __shared__ int   s_src[2][TILE_EDGES];
  __shared__ int   s_dst[2][TILE_EDGES];

  const int tid = threadIdx.x;
  if (tid < N_GRAPHS) { s_sum[tid] = 0.0f; s_cnt[tid] = 0.0f; }
  __syncthreads();

  const int n_tiles = n_edges / TILE_EDGES;
  const int stride  = gridDim.x;
  const int* dsts   = eidx + n_edges;

  int t   = blockIdx.x;
  int buf = 0;

#if USE_ASYNC
  if (t < n_tiles) {  // prologue: prefetch first tile into buffer 0
    const int base = t * TILE_EDGES + tid * 4;
    async_b128(eidx + base, &s_src[0][tid * 4]);
    async_b128(dsts + base, &s_dst[0][tid * 4]);
  }
#endif

  for (; t < n_tiles; t += stride) {
    int4 s, d;
#if USE_ASYNC
    const int nt = t + stride;
    if (nt < n_tiles) {  // prefetch next tile into the other buffer
      const int nb = nt * TILE_EDGES + tid * 4;
      async_b128(eidx + nb, &s_src[buf ^ 1][tid * 4]);
      async_b128(dsts + nb, &s_dst[buf ^ 1][tid * 4]);
      __builtin_amdgcn_s_wait_asynccnt(2);  // current tile's 2 ops retired
    } else {
      __builtin_amdgcn_s_wait_asynccnt(0);
    }
    __syncthreads();                         // all waves' copies visible
    s = ((const int4*)s_src[buf])[tid];      // ds_load_b128
    d = ((const int4*)s_dst[buf])[tid];
#else
    const int base = t * TILE_EDGES + tid * 4;
    s = *(const int4*)(eidx + base);         // global_load_b128 fallback
    d = *(const int4*)(dsts + base);
#endif
    edge_accum(pos, batch, s.x, d.x, s_sum, s_cnt);
    edge_accum(pos, batch, s.y, d.y, s_sum, s_cnt);
    edge_accum(pos, batch, s.z, d.z, s_sum, s_cnt);
    edge_accum(pos, batch, s.w, d.w, s_sum, s_cnt);
#if USE_ASYNC
    __syncthreads();  // protect buf before next iteration overwrites it
#endif
    buf ^= 1;
  }

  // tail edges (none for E = 3.2M, kept for generality): block 0 handles them
  if (blockIdx.x == 0) {
    for (int e = n_tiles * TILE_EDGES + tid; e < n_edges; e += BLOCK)
      edge_accum(pos, batch, eidx[e], dsts[e], s_sum, s_cnt);
  }

  __syncthreads();
  // one global fp32 atomic per bin per block
  if (tid < N_GRAPHS) {
    if (s_sum[tid] != 0.0f) unsafeAtomicAdd(&ws[tid], s_sum[tid]);
    if (s_cnt[tid] != 0.0f) unsafeAtomicAdd(&ws[N_GRAPHS + tid], s_cnt[tid]);
  }
}

__global__ void edgevar_zero_ws(float* ws) {
  int i = blockIdx.x * blockDim.x + threadIdx.x;
  if (i < 2 * N_GRAPHS) ws[i] = 0.0f;
}

__global__ void edgevar_finalize(const float* __restrict__ ws,
                                 float* __restrict__ out) {
  __shared__ float red[N_GRAPHS];
  const int g = threadIdx.x;  // 128 threads
  red[g] = ws[g] / fmaxf(ws[N_GRAPHS + g], 1.0f);
  __syncthreads();
  for (int off = N_GRAPHS / 2; off > 0; off >>= 1) {
    if (g < off) red[g] += red[g + off];
    __syncthreads();
  }
  if (g == 0) out[0] = red[0] * (1.0f / (float)N_GRAPHS);
}

extern "C" void kernel_launch(void* const* d_in, const int* in_sizes, int n_in,
                              void* d_out, int out_size, void* d_ws, size_t ws_size,
                              hipStream_t stream) {
  const float* pos   = (const float*)d_in[0];   // node_pos [N,2] fp32
  const int*   eidx  = (const int*)d_in[1];     // raw_edge_index [2,E]
  const int*   batch = (const int*)d_in[2];     // batch [N]
  float*       ws    = (float*)d_ws;            // 256 floats: sums | counts
  float*       out   = (float*)d_out;           // scalar fp32

  const int n_edges = in_sizes[1] / 2;
  const int n_tiles = n_edges / TILE_EDGES;

  edgevar_zero_ws<<<1, 256, 0, stream>>>(ws);

  // enough blocks to fill the chip, enough tiles/block (~5) to amortize the
  // async double-buffer pipeline
  int grid = n_tiles < 640 ? (n_tiles > 0 ? n_tiles : 1) : 640;
  edgevar_main<<<grid, BLOCK, 0, stream>>>(pos, eidx, batch, ws, n_edges);

  edgevar_finalize<<<1, N_GRAPHS, 0, stream>>>(ws, out);
}